// CrossAttentionBlock_91096256348202
// MI455X (gfx1250) — compile-verified
//
#include <hip/hip_runtime.h>
#include <hip/hip_bf16.h>

typedef __attribute__((ext_vector_type(2))) float v2f;
typedef __attribute__((ext_vector_type(8))) float v8f;

#define BATCH 32
#define TDIM  768
#define CDIM  448
#define EDIM  256
#define NPIX  3136   // 56*56

// Async global->LDS copy helpers (gfx1250 ASYNCcnt path). LDS byte address is
// the low 32 bits of the generic pointer (addr[31:0] per LDS aperture rules).
__device__ __forceinline__ void async_copy_b32(const float* gsrc, float* ldst) {
    unsigned lds = (unsigned)(unsigned long long)ldst;
    asm volatile("global_load_async_to_lds_b32 %0, %1, off"
                 :: "v"(lds), "v"(gsrc) : "memory");
}
__device__ __forceinline__ void async_copy_b128(const float* gsrc, float* ldst) {
    unsigned lds = (unsigned)(unsigned long long)ldst;
    asm volatile("global_load_async_to_lds_b128 %0, %1, off"
                 :: "v"(lds), "v"(gsrc) : "memory");
}
__device__ __forceinline__ void wait_async0() {
    asm volatile("s_wait_asynccnt 0x0" ::: "memory");
}

// ---------------------------------------------------------------------------
// Small GEMM on the WMMA path:  C[32 x N] = A[32 x K] * B + bias
// One wave per 16x16 tile; K consumed 4/step via V_WMMA_F32_16X16X4_F32.
// TRANSB=1 : B stored [N x K]   (q_w, v_w, o_w)
// TRANSB=0 : B stored [K x N]   (k_w)
// transB/bias are template params so the inner loop is branch-free.
// ---------------------------------------------------------------------------
template <int TRANSB, int HASBIAS>
__global__ void wmma_gemm_f32(const float* __restrict__ A, int lda,
                              const float* __restrict__ B, int ldb,
                              const float* __restrict__ bias,
                              float* __restrict__ C, int ldc, int K) {
    const int lane = threadIdx.x & 31;
    const int half = lane >> 4;        // 0: K=0,1  1: K=2,3 (frag layout)
    const int l15  = lane & 15;
    const int tm   = blockIdx.y;
    const int tn   = blockIdx.x;
    const int am   = tm * 16 + l15;    // A row for this lane
    const int bn   = tn * 16 + l15;    // B col for this lane

    const float* __restrict__ Ap = A + (size_t)am * lda + half * 2;
    const float* __restrict__ Bp = TRANSB ? (B + (size_t)bn * ldb + half * 2)
                                          : (B + (size_t)(half * 2) * ldb + bn);

    v8f acc = {};
#pragma unroll 4
    for (int k0 = 0; k0 < K; k0 += 4) {
        v2f a, b;
        a.x = Ap[k0];
        a.y = Ap[k0 + 1];
        if (TRANSB) {
            b.x = Bp[k0];
            b.y = Bp[k0 + 1];
        } else {
            b.x = Bp[(size_t)k0 * ldb];
            b.y = Bp[(size_t)(k0 + 1) * ldb];
        }
        acc = __builtin_amdgcn_wmma_f32_16x16x4_f32(
            false, a, false, b, (short)0, acc, false, false);
    }

    const int col = tn * 16 + l15;
    const float bv = HASBIAS ? bias[col] : 0.0f;
#pragma unroll
    for (int i = 0; i < 8; ++i) {
        const int row = tm * 16 + half * 8 + i;   // C/D layout: VGPR i -> M=i (+8 hi half)
        C[(size_t)row * ldc + col] = acc[i] + bv;
    }
}

// qkb[b] = sum_e q[b,e] * k_b[e]   (folded key bias)
__global__ void qkb_kernel(const float* __restrict__ q,
                           const float* __restrict__ k_b,
                           float* __restrict__ qkb) {
    const int b = blockIdx.x;
    const int lane = threadIdx.x;
    float s = 0.0f;
    for (int e = lane; e < EDIM; e += 32) s += q[b * EDIM + e] * k_b[e];
    for (int off = 16; off; off >>= 1) s += __shfl_down(s, off);
    if (lane == 0) qkb[b] = s;
}

// scores[b,n] = ( sum_c qk[b,c]*img[b,c,n] + qkb[b] ) * 1/sqrt(E)
// Pass 1 over the image. qk row staged into LDS with async copies.
__global__ void scores_kernel(const float* __restrict__ img,
                              const float* __restrict__ qk,
                              const float* __restrict__ qkb,
                              float* __restrict__ scores) {
    const int b = blockIdx.y;
    __shared__ float sqk[CDIM];

    {   // async global->LDS: 448 floats, 112 b128 transfers across 256 lanes
        const float* g = qk + (size_t)b * CDIM;
        if (threadIdx.x < CDIM / 4)
            async_copy_b128(g + threadIdx.x * 4, &sqk[threadIdx.x * 4]);
        wait_async0();
    }
    __syncthreads();

    const int n0 = blockIdx.x * 1024 + threadIdx.x * 4;
    if (n0 >= NPIX) return;

    const float* base = img + (size_t)b * CDIM * NPIX + n0;
    float4 acc = {0.f, 0.f, 0.f, 0.f};
#pragma unroll 4
    for (int c = 0; c < CDIM; ++c) {
        const float  w = sqk[c];
        const float4 x = *(const float4*)(base + (size_t)c * NPIX);
        acc.x += w * x.x; acc.y += w * x.y; acc.z += w * x.z; acc.w += w * x.w;
    }
    const float bias  = qkb[b];
    const float scale = 0.0625f;   // 1/sqrt(256)
    float4 r;
    r.x = (acc.x + bias) * scale;  r.y = (acc.y + bias) * scale;
    r.z = (acc.z + bias) * scale;  r.w = (acc.w + bias) * scale;
    *(float4*)(scores + (size_t)b * NPIX + n0) = r;
}

// Per-batch softmax numerator: scores -> exp(scores - max) in place; sums[b] = sum.
__global__ void softmax_kernel(float* __restrict__ scores, float* __restrict__ sums) {
    const int b = blockIdx.x;
    const int t = threadIdx.x;
    float* s = scores + (size_t)b * NPIX;

    __shared__ float red[256];
    float m = -1e30f;
    for (int n = t; n < NPIX; n += 256) m = fmaxf(m, s[n]);
    red[t] = m; __syncthreads();
    for (int off = 128; off; off >>= 1) {
        if (t < off) red[t] = fmaxf(red[t], red[t + off]);
        __syncthreads();
    }
    m = red[0]; __syncthreads();

    float sum = 0.0f;
    for (int n = t; n < NPIX; n += 256) {
        const float e = __expf(s[n] - m);
        s[n] = e;
        sum += e;
    }
    red[t] = sum; __syncthreads();
    for (int off = 128; off; off >>= 1) {
        if (t < off) red[t] += red[t + off];
        __syncthreads();
    }
    if (t == 0) sums[b] = red[0];
}

// ctx[b,c] = ( sum_n p[b,n]*img[b,c,n] ) / sums[b]
// One wave per (b,c); the 12.5KB attention row p[b,:] is staged once per block
// into LDS via async b128 copies and shared by all 8 waves.
__global__ void ctx_kernel(const float* __restrict__ img,
                           const float* __restrict__ p,
                           const float* __restrict__ sums,
                           float* __restrict__ ctx) {
    __shared__ float sp[NPIX];
    const int b    = blockIdx.x / (CDIM / 8);   // CDIM%8==0 -> one b per block
    const int cgrp = blockIdx.x % (CDIM / 8);

    {   // async global->LDS: 3136 floats = 784 b128 transfers
        const float* g = p + (size_t)b * NPIX;
        for (int i = threadIdx.x; i < NPIX / 4; i += 256)
            async_copy_b128(g + i * 4, &sp[i * 4]);
        wait_async0();
    }
    __syncthreads();

    const int c    = cgrp * 8 + (threadIdx.x >> 5);
    const int lane = threadIdx.x & 31;
    const float* row = img + ((size_t)b * CDIM + c) * NPIX;

    float acc = 0.0f;
    for (int n = lane * 4; n < NPIX; n += 128) {
        const float4 x = *(const float4*)(row + n);
        const float4 w = *(const float4*)(&sp[n]);
        acc += x.x * w.x + x.y * w.y + x.z * w.z + x.w * w.w;
    }
    for (int off = 16; off; off >>= 1) acc += __shfl_down(acc, off);
    if (lane == 0) ctx[b * CDIM + c] = acc / sums[b];
}

// out[b,c,:,:] = img[b,c,:,:] + outvec[b,c]
__global__ void add_kernel(const float* __restrict__ img,
                           const float* __restrict__ outvec,
                           float* __restrict__ out) {
    const int bc = blockIdx.y;                       // b*CDIM + c
    const int i4 = blockIdx.x * 256 + threadIdx.x;   // float4 index within row
    if (i4 >= NPIX / 4) return;
    const float  o    = outvec[bc];
    const size_t base = (size_t)bc * NPIX + (size_t)i4 * 4;
    const float4 x = *(const float4*)(img + base);
    float4 r = {x.x + o, x.y + o, x.z + o, x.w + o};
    *(float4*)(out + base) = r;
}

extern "C" void kernel_launch(void* const* d_in, const int* in_sizes, int n_in,
                              void* d_out, int out_size, void* d_ws, size_t ws_size,
                              hipStream_t stream) {
    const float* text = (const float*)d_in[0];   // [32,768]
    const float* img  = (const float*)d_in[1];   // [32,448,56,56]
    const float* q_w  = (const float*)d_in[2];   // [256,768]
    const float* q_b  = (const float*)d_in[3];   // [256]
    const float* k_w  = (const float*)d_in[4];   // [256,448]
    const float* k_b  = (const float*)d_in[5];   // [256]
    const float* v_w  = (const float*)d_in[6];   // [256,448]
    const float* v_b  = (const float*)d_in[7];   // [256]
    const float* o_w  = (const float*)d_in[8];   // [448,256]
    const float* o_b  = (const float*)d_in[9];   // [448]
    float* out = (float*)d_out;

    float* ws      = (float*)d_ws;
    float* q       = ws;                     // 32*256
    float* qk      = q + BATCH * EDIM;       // 32*448
    float* qkb     = qk + BATCH * CDIM;      // 32
    float* sums    = qkb + BATCH;            // 32
    float* ctx     = sums + BATCH;           // 32*448
    float* pooled  = ctx + BATCH * CDIM;     // 32*256
    float* outvec  = pooled + BATCH * EDIM;  // 32*448
    float* scores  = outvec + BATCH * CDIM;  // 32*3136

    // 1) q = text @ q_w^T + q_b                       [32,256]  (WMMA)
    wmma_gemm_f32<1, 1><<<dim3(EDIM / 16, 2), 32, 0, stream>>>(
        text, TDIM, q_w, TDIM, q_b, q, EDIM, TDIM);
    // 2) qk = q @ k_w                                 [32,448]  (WMMA)
    wmma_gemm_f32<0, 0><<<dim3(CDIM / 16, 2), 32, 0, stream>>>(
        q, EDIM, k_w, CDIM, nullptr, qk, CDIM, EDIM);
    // 3) qkb = q . k_b                                [32]
    qkb_kernel<<<BATCH, 32, 0, stream>>>(q, k_b, qkb);
    // 4) scores pass over image                       [32,3136]
    scores_kernel<<<dim3((NPIX + 1023) / 1024, BATCH), 256, 0, stream>>>(
        img, qk, qkb, scores);
    // 5) softmax numerator + per-batch sums
    softmax_kernel<<<BATCH, 256, 0, stream>>>(scores, sums);
    // 6) ctx = (attn . img) over spatial              [32,448]
    ctx_kernel<<<(BATCH * CDIM) / 8, 256, 0, stream>>>(img, scores, sums, ctx);
    // 7) pooled = ctx @ v_w^T + v_b                   [32,256]  (WMMA)
    wmma_gemm_f32<1, 1><<<dim3(EDIM / 16, 2), 32, 0, stream>>>(
        ctx, CDIM, v_w, CDIM, v_b, pooled, EDIM, CDIM);
    // 8) outvec = pooled @ o_w^T + o_b                [32,448]  (WMMA)
    wmma_gemm_f32<1, 1><<<dim3(CDIM / 16, 2), 32, 0, stream>>>(
        pooled, EDIM, o_w, EDIM, o_b, outvec, CDIM, EDIM);
    // 9) out = img + outvec broadcast                 [32,448,56,56]
    add_kernel<<<dim3((NPIX / 4 + 255) / 256, BATCH * CDIM), 256, 0, stream>>>(
        img, outvec, out);
}